// Method_GCN_39419209843175
// MI455X (gfx1250) — compile-verified
//
#include <hip/hip_runtime.h>
#include <hip/hip_bf16.h>

typedef __attribute__((ext_vector_type(2))) float v2f;
typedef __attribute__((ext_vector_type(8))) float v8f;

#define N_NODES 100000
#define N_EDGES 3200000
#define IN_DIM  512
#define HID_DIM 256
#define OUT_DIM 64

__global__ void zero_kernel(float* __restrict__ p, long n) {
  long i = (long)blockIdx.x * blockDim.x + threadIdx.x;
  long stride = (long)gridDim.x * blockDim.x;
  for (; i < n; i += stride) p[i] = 0.0f;
}

// Bp[k2*N + n] = (B[2*k2][n], B[2*k2+1][n]) : K-pair interleave so each WMMA
// B operand is one contiguous 8-byte load.
__global__ __launch_bounds__(256) void interleave_pairs(
    const float* __restrict__ B, float2* __restrict__ Bp, int K, int N) {
  int i = blockIdx.x * blockDim.x + threadIdx.x;
  int total = (K / 2) * N;
  if (i >= total) return;
  int k2 = i / N, n = i - k2 * N;
  float2 v;
  v.x = B[(2 * k2) * N + n];
  v.y = B[(2 * k2 + 1) * N + n];
  Bp[i] = v;
}

// C[M,N] = op(A[M,K]) @ B[K,N] with fp32 WMMA 16x16x4, B pre-interleaved.
// Each wave: 16-row strip x 64-col panel (4 accumulators). 4 waves/block.
// Software-pipelined: next k-step's A/B loads issued before current WMMAs.
template<int K, int RELU>
__global__ __launch_bounds__(128) void gemm_wmma_f32(
    const float* __restrict__ A, const float2* __restrict__ Bp,
    float* __restrict__ C, int M, int N) {
  const int lane = threadIdx.x & 31;
  const int rt   = blockIdx.x * 4 + (threadIdx.x >> 5);   // 16-row tile index
  if (rt * 16 >= M) return;                                // wave-uniform exit
  const int ct   = blockIdx.y;                             // 64-col tile index
  const int half = lane >> 4;                              // 0: K0/K1, 1: K2/K3
  const int l15  = lane & 15;
  const int kb   = half * 2;
  const int colbase = ct * 64;

  const float*  Arow  = A  + (long)(rt * 16 + l15) * K;
  // row index in Bp for k-step k is (k+kb)/2 ; starts at kb/2 = half, advances by 2
  const float2* Brow0 = Bp + (long)half * N + colbase + l15;

  v8f c0 = {}, c1 = {}, c2 = {}, c3 = {};

  // prologue: loads for k = 0
  v2f a_cur;
  a_cur.x = Arow[kb];
  a_cur.y = Arow[kb + 1];
  float2 b_cur0 = Brow0[0];
  float2 b_cur1 = Brow0[16];
  float2 b_cur2 = Brow0[32];
  float2 b_cur3 = Brow0[48];

  for (int k = 4; k < K; k += 4) {
    // issue next iteration's loads first
    v2f a_nxt;
    a_nxt.x = Arow[k + kb];
    a_nxt.y = Arow[k + kb + 1];
    const float2* Bn = Brow0 + (long)(k >> 1) * N;
    float2 b_nxt0 = Bn[0];
    float2 b_nxt1 = Bn[16];
    float2 b_nxt2 = Bn[32];
    float2 b_nxt3 = Bn[48];

    if (RELU) { a_cur.x = fmaxf(a_cur.x, 0.0f); a_cur.y = fmaxf(a_cur.y, 0.0f); }
    v2f b;
    b.x = b_cur0.x; b.y = b_cur0.y;
    c0 = __builtin_amdgcn_wmma_f32_16x16x4_f32(false, a_cur, false, b, (short)0, c0, false, false);
    b.x = b_cur1.x; b.y = b_cur1.y;
    c1 = __builtin_amdgcn_wmma_f32_16x16x4_f32(false, a_cur, false, b, (short)0, c1, false, false);
    b.x = b_cur2.x; b.y = b_cur2.y;
    c2 = __builtin_amdgcn_wmma_f32_16x16x4_f32(false, a_cur, false, b, (short)0, c2, false, false);
    b.x = b_cur3.x; b.y = b_cur3.y;
    c3 = __builtin_amdgcn_wmma_f32_16x16x4_f32(false, a_cur, false, b, (short)0, c3, false, false);

    a_cur = a_nxt;
    b_cur0 = b_nxt0; b_cur1 = b_nxt1; b_cur2 = b_nxt2; b_cur3 = b_nxt3;
  }

  // epilogue: last k-step
  if (RELU) { a_cur.x = fmaxf(a_cur.x, 0.0f); a_cur.y = fmaxf(a_cur.y, 0.0f); }
  {
    v2f b;
    b.x = b_cur0.x; b.y = b_cur0.y;
    c0 = __builtin_amdgcn_wmma_f32_16x16x4_f32(false, a_cur, false, b, (short)0, c0, false, false);
    b.x = b_cur1.x; b.y = b_cur1.y;
    c1 = __builtin_amdgcn_wmma_f32_16x16x4_f32(false, a_cur, false, b, (short)0, c1, false, false);
    b.x = b_cur2.x; b.y = b_cur2.y;
    c2 = __builtin_amdgcn_wmma_f32_16x16x4_f32(false, a_cur, false, b, (short)0, c2, false, false);
    b.x = b_cur3.x; b.y = b_cur3.y;
    c3 = __builtin_amdgcn_wmma_f32_16x16x4_f32(false, a_cur, false, b, (short)0, c3, false, false);
  }

  // C/D layout: VGPR r -> row (r + 8*half), N = lane&15 within each 16-col tile
  for (int r = 0; r < 8; ++r) {
    long row = (long)rt * 16 + r + 8 * half;
    float* Crow = C + row * (long)N + colbase + l15;
    Crow[0]  = c0[r];
    Crow[16] = c1[r];
    Crow[32] = c2[r];
    Crow[48] = c3[r];
  }
}

// Edge-parallel SpMM: out[dst] += w * sup[src], F = (1<<LOGF4) float4 chunks per row.
// Consecutive threads take consecutive float4 chunks of one edge -> coalesced gather.
template<int LOGF4>
__global__ __launch_bounds__(256) void spmm_atomic(
    const int* __restrict__ src, const int* __restrict__ dst,
    const float* __restrict__ w, const float* __restrict__ sup,
    float* __restrict__ out) {
  const long total = (long)N_EDGES << LOGF4;
  long i = (long)blockIdx.x * blockDim.x + threadIdx.x;
  const long stride = (long)gridDim.x * blockDim.x;
  for (; i < total; i += stride) {
    const long e = i >> LOGF4;
    const int  c = (int)(i & ((1 << LOGF4) - 1));
    const int s = src[e];
    const int d = dst[e];
    const float ww = w[e];
    const float4 v = ((const float4*)(sup + ((long)s << (LOGF4 + 2))))[c];
    float* orow = out + ((long)d << (LOGF4 + 2)) + c * 4;
    unsafeAtomicAdd(orow + 0, v.x * ww);
    unsafeAtomicAdd(orow + 1, v.y * ww);
    unsafeAtomicAdd(orow + 2, v.z * ww);
    unsafeAtomicAdd(orow + 3, v.w * ww);
  }
}

// In-place row softmax over 64 columns; one wave32 per row, 2 cols/lane.
__global__ __launch_bounds__(256) void softmax64(float* __restrict__ io, int nrows) {
  const int row = blockIdx.x * (blockDim.x >> 5) + (threadIdx.x >> 5);
  if (row >= nrows) return;
  const int lane = threadIdx.x & 31;
  float* p = io + (long)row * 64;
  float v0 = p[lane];
  float v1 = p[lane + 32];
  float m = fmaxf(v0, v1);
  #pragma unroll
  for (int off = 16; off; off >>= 1) m = fmaxf(m, __shfl_xor(m, off, 32));
  float e0 = __expf(v0 - m), e1 = __expf(v1 - m);
  float s = e0 + e1;
  #pragma unroll
  for (int off = 16; off; off >>= 1) s += __shfl_xor(s, off, 32);
  const float inv = 1.0f / s;
  p[lane]      = e0 * inv;
  p[lane + 32] = e1 * inv;
}

extern "C" void kernel_launch(void* const* d_in, const int* in_sizes, int n_in,
                              void* d_out, int out_size, void* d_ws, size_t ws_size,
                              hipStream_t stream) {
  const float* x  = (const float*)d_in[0];   // [100000, 512]
  const int*   ei = (const int*)d_in[1];     // [2, E] row-major: src then dst
  const float* ew = (const float*)d_in[2];   // [E]
  const float* W1 = (const float*)d_in[3];   // [512, 256]
  const float* W2 = (const float*)d_in[4];   // [256, 64]
  float* out = (float*)d_out;                // [100000, 64]

  float*  sup1 = (float*)d_ws;                        // 100000*256 floats
  float*  h    = sup1 + (long)N_NODES * HID_DIM;      // 100000*256 floats
  float*  sup2 = h    + (long)N_NODES * HID_DIM;      // 100000*64  floats
  float2* Bp1  = (float2*)(sup2 + (long)N_NODES * OUT_DIM);          // 256*256 pairs
  float2* Bp2  = Bp1 + (IN_DIM / 2) * HID_DIM;                       // 128*64  pairs
  const int* srcv = ei;
  const int* dstv = ei + N_EDGES;

  const int ROW_TILES = N_NODES / 16;  // 6250

  // pre-pass: interleave W1/W2 K-pairs; zero the layer-1 accumulator
  interleave_pairs<<<((IN_DIM / 2) * HID_DIM + 255) / 256, 256, 0, stream>>>(W1, Bp1, IN_DIM, HID_DIM);
  interleave_pairs<<<((HID_DIM / 2) * OUT_DIM + 255) / 256, 256, 0, stream>>>(W2, Bp2, HID_DIM, OUT_DIM);
  zero_kernel<<<2048, 256, 0, stream>>>(h, (long)N_NODES * HID_DIM);

  // sup1 = x @ W1  (fp32 WMMA)
  dim3 g1((ROW_TILES + 3) / 4, HID_DIM / 64);
  gemm_wmma_f32<IN_DIM, 0><<<g1, 128, 0, stream>>>(x, Bp1, sup1, N_NODES, HID_DIM);

  // h[dst] += w * sup1[src]   (F=256 -> 64 float4 chunks, LOGF4=6)
  spmm_atomic<6><<<800000, 256, 0, stream>>>(srcv, dstv, ew, sup1, h);

  // sup2 = relu(h) @ W2  (ReLU fused into A-load)
  dim3 g2((ROW_TILES + 3) / 4, OUT_DIM / 64);
  gemm_wmma_f32<HID_DIM, 1><<<g2, 128, 0, stream>>>(h, Bp2, sup2, N_NODES, OUT_DIM);

  // out = 0, then out[dst] += w * sup2[src]  (F=64 -> 16 float4 chunks, LOGF4=4)
  zero_kernel<<<1024, 256, 0, stream>>>(out, (long)N_NODES * OUT_DIM);
  spmm_atomic<4><<<200000, 256, 0, stream>>>(srcv, dstv, ew, sup2, out);

  // row-wise softmax in place
  softmax64<<<(N_NODES + 7) / 8, 256, 0, stream>>>(out, N_NODES);
}